// QueryPairwiseRankLoss_15075335209002
// MI455X (gfx1250) — compile-verified
//
#include <hip/hip_runtime.h>

typedef float v2f __attribute__((ext_vector_type(2)));
typedef float v8f __attribute__((ext_vector_type(8)));

#define GROUP_SIZE 128

// ---------------------------------------------------------------------------
// One workgroup (128 threads = 4 wave32) per group.
// Stage scores+labels in LDS, each thread owns row i, loops over j.
// Wave-level reduction of the 32 lane partials uses V_WMMA_F32_16X16X4_F32
// with an all-ones B matrix:
//   A layout (16x4 f32): a[0] = lane partial lands at (M=lane%16, K= lane<16?0:2)
//   With B == ones:      D[m,n] = part_m + part_{m+16}
//   C/D layout: lane 0 VGPRs 0..7 hold M=0..7, lane 16 VGPRs 0..7 hold M=8..15
//   => sum(D rows) via lane0 + lane16 gives the full 32-lane sum.
// ---------------------------------------------------------------------------
__global__ __launch_bounds__(GROUP_SIZE)
void rank_loss_kernel(const float* __restrict__ scores,
                      const int*   __restrict__ labels,
                      float* __restrict__ pg_out,     // [ng] per-group loss (0 if invalid)
                      float* __restrict__ valid_out)  // [ng] 1.0 if group has valid pairs
{
    __shared__ float s_s[GROUP_SIZE];
    __shared__ int   s_l[GROUP_SIZE];
    __shared__ float red[8];

    const int g    = blockIdx.x;
    const int t    = threadIdx.x;
    const int base = g * GROUP_SIZE;

    // gfx1250 prefetch path (lowers to global_prefetch_b8)
    __builtin_prefetch(scores + base + t, 0, 0);
    __builtin_prefetch(labels + base + t, 0, 0);

    // Stage group into LDS (broadcast-friendly inner-loop reads).
    s_s[t] = scores[base + t];
    s_l[t] = labels[base + t];
    __syncthreads();

    const float si = s_s[t];
    const int   li = s_l[t];

    float acc = 0.0f;
    int   cnt = 0;

#pragma unroll 8
    for (int j = 0; j < GROUP_SIZE; ++j) {
        const float x  = s_s[j] - si;                 // s_j - s_i  (== -diff)
        const int   m  = (li > s_l[j]);
        // numerically stable softplus(x) = max(x,0) + log1p(exp(-|x|))
        const float sp = fmaxf(x, 0.0f) + log1pf(__expf(-fabsf(x)));
        acc += m ? sp : 0.0f;                         // v_cndmask, no EXEC change
        cnt += m;
    }

    // ---- wave32 reduction via WMMA (EXEC is all 1s here) ----
    v2f a;  a[0] = acc;         a[1] = 0.0f;
    v2f c;  c[0] = (float)cnt;  c[1] = 0.0f;
    v2f b;  b[0] = 1.0f;        b[1] = 1.0f;          // all-ones B (layout-independent)

    v8f dA = {};
    dA = __builtin_amdgcn_wmma_f32_16x16x4_f32(false, a, false, b, (short)0, dA, false, false);
    v8f dC = {};
    dC = __builtin_amdgcn_wmma_f32_16x16x4_f32(false, c, false, b, (short)0, dC, false, false);

    float pA = dA[0] + dA[1] + dA[2] + dA[3] + dA[4] + dA[5] + dA[6] + dA[7];
    float pC = dC[0] + dC[1] + dC[2] + dC[3] + dC[4] + dC[5] + dC[6] + dC[7];

    // lane 0 holds sum of M=0..7, lane 16 holds sum of M=8..15
    const float waveAcc = __shfl(pA, 0, 32) + __shfl(pA, 16, 32);
    const float waveCnt = __shfl(pC, 0, 32) + __shfl(pC, 16, 32);

    const int wave = t >> 5;
    if ((t & 31) == 0) {
        red[wave * 2 + 0] = waveAcc;
        red[wave * 2 + 1] = waveCnt;
    }
    __syncthreads();

    if (t == 0) {
        const float gA = red[0] + red[2] + red[4] + red[6];
        const float gC = red[1] + red[3] + red[5] + red[7];
        const bool valid = (gC > 0.0f);
        pg_out[g]    = valid ? (gA / fmaxf(gC, 1.0f)) : 0.0f;
        valid_out[g] = valid ? 1.0f : 0.0f;
    }
}

// ---------------------------------------------------------------------------
// Deterministic final reduction: fixed-order strided sums + LDS tree.
// ---------------------------------------------------------------------------
__global__ __launch_bounds__(256)
void final_reduce_kernel(const float* __restrict__ pg,
                         const float* __restrict__ vd,
                         int ng,
                         float* __restrict__ out)
{
    __shared__ float sh[256];
    __shared__ float sv[256];
    const int t = threadIdx.x;

    float a = 0.0f, v = 0.0f;
    for (int i = t; i < ng; i += 256) {
        a += pg[i];
        v += vd[i];
    }
    sh[t] = a;
    sv[t] = v;
    __syncthreads();

    for (int s = 128; s > 0; s >>= 1) {
        if (t < s) {
            sh[t] += sh[t + s];
            sv[t] += sv[t + s];
        }
        __syncthreads();
    }

    if (t == 0) {
        out[0] = (sv[0] > 0.0f) ? (sh[0] / sv[0]) : 0.0f;
    }
}

extern "C" void kernel_launch(void* const* d_in, const int* in_sizes, int n_in,
                              void* d_out, int out_size, void* d_ws, size_t ws_size,
                              hipStream_t stream) {
    const float* scores = (const float*)d_in[0];
    const int*   labels = (const int*)d_in[1];
    // d_in[2] is group_size (==128), compile-time constant here.

    const int n  = in_sizes[0];
    const int ng = n / GROUP_SIZE;

    float* pg = (float*)d_ws;        // [ng]
    float* vd = pg + ng;             // [ng]

    rank_loss_kernel<<<ng, GROUP_SIZE, 0, stream>>>(scores, labels, pg, vd);
    final_reduce_kernel<<<1, 256, 0, stream>>>(pg, vd, ng, (float*)d_out);
}